// INSIntegrationLoss_25993142075962
// MI455X (gfx1250) — compile-verified
//
#include <hip/hip_runtime.h>

// INS integration loss for MI455X (gfx1250, wave32).
// Grid: 256 blocks (one per batch), 256 threads (8 waves) per block.
// Quaternion attitude: hierarchical non-commutative prefix product
//   (8 serial steps/lane -> wave32 shfl scan -> LDS cross-wave -> chunk carry).
// Velocity/position: double prefix sum done per 16-step tile with
//   V_WMMA_F32_16X16X4_F32:  sv = L*X, sp = T*X  (L = lower-tri ones,
//   T[m,k] = m-k+1), carries combined with the affine segment rule.
// B operand zero-padding is done via a zero LDS row (no exec-masked loads).

#define B_      256
#define S_      8192
#define NSTEP   (S_ - 1)      // 8191 integration steps
#define DTg     0.01f
#define DT2g    (DTg * DTg)
#define EPSg    1e-12f
#define CHUNK   2048
#define NCHUNK  4
#define LSTEPS  8             // steps per thread per chunk (2048/256)
#define BLOCK   256
#define WAVES   8

typedef __attribute__((ext_vector_type(2))) float v2f;
typedef __attribute__((ext_vector_type(8))) float v8f;

struct Quat { float w, x, y, z; };

__device__ __forceinline__ Quat qmul(const Quat& a, const Quat& b) {
    Quat r;
    r.w = a.w*b.w - a.x*b.x - a.y*b.y - a.z*b.z;
    r.x = a.w*b.x + a.x*b.w + a.y*b.z - a.z*b.y;
    r.y = a.w*b.y - a.x*b.z + a.y*b.w + a.z*b.x;
    r.z = a.w*b.z + a.x*b.y - a.y*b.x + a.z*b.w;
    return r;
}

__device__ __forceinline__ Quat qnorm(const Quat& q) {
    float n = sqrtf(q.w*q.w + q.x*q.x + q.y*q.y + q.z*q.z) + EPSg;
    float inv = 1.0f / n;
    Quat r; r.w = q.w*inv; r.x = q.x*inv; r.y = q.y*inv; r.z = q.z*inv;
    return r;
}

__device__ __forceinline__ Quat qshfl_up(const Quat& q, int off) {
    Quat r;
    r.w = __shfl_up(q.w, off, 32);
    r.x = __shfl_up(q.x, off, 32);
    r.y = __shfl_up(q.y, off, 32);
    r.z = __shfl_up(q.z, off, 32);
    return r;
}

__global__ void ins_zero_out_kernel(float* out) {
    if (threadIdx.x == 0) out[0] = 0.0f;
}

__global__ __launch_bounds__(BLOCK)
void ins_loss_kernel(const float* __restrict__ predicted,
                     const float* __restrict__ gt,
                     const float* __restrict__ us,
                     float* __restrict__ out) {
    const int b    = blockIdx.x;
    const int tid  = threadIdx.x;
    const int lane = tid & 31;
    const int wid  = tid >> 5;
    const int n    = lane & 15;   // WMMA column index this lane owns in B/D

    // rows 0..2: NED accel x/y/z for the chunk; row 3: permanent zeros so
    // lanes with n >= 3 read harmless zero columns with NO divergent control.
    __shared__ float s_a[4][CHUNK];
    __shared__ float s_waveQ[WAVES][4];
    __shared__ float s_preQ[WAVES][4];
    __shared__ float s_qc[2][4];
    __shared__ float s_vc[2][3];
    __shared__ float s_pc[2][3];
    __shared__ float s_wSv[WAVES][3], s_wSp[WAVES][3];
    __shared__ float s_wVin[WAVES][3], s_wPin[WAVES][3];
    __shared__ float s_red[WAVES];

    const float invVP = 1.0f / ((float)B_ * (float)S_ * 3.0f);
    const float invQ  = 1.0f / ((float)B_ * (float)S_ * 4.0f);

    const float* predB = predicted + (size_t)b * S_ * 6;
    const float* usB   = us        + (size_t)b * S_ * 3;
    const float* gtB   = gt        + (size_t)b * S_ * 10;

    float acc = 0.0f;

    // zero row 3 once (never rewritten)
#pragma unroll
    for (int k = 0; k < LSTEPS; ++k) s_a[3][tid * LSTEPS + k] = 0.0f;

    if (tid == 0) {
        s_qc[0][0] = 1.f; s_qc[0][1] = 0.f; s_qc[0][2] = 0.f; s_qc[0][3] = 0.f;
        for (int c = 0; c < 3; ++c) { s_vc[0][c] = 0.f; s_pc[0][c] = 0.f; }
        // t = 0 terms: q0 = (1,0,0,0), v0 = p0 = 0
        float dw = 1.0f - gtB[6];
        acc += (dw*dw + gtB[7]*gtB[7] + gtB[8]*gtB[8] + gtB[9]*gtB[9]) * invQ;
        for (int c = 0; c < 3; ++c)
            acc += (gtB[c]*gtB[c] + gtB[3 + c]*gtB[3 + c]) * invVP;
    }

    // Constant A operands for the scan matrices (per ISA 16x4 fp32 A layout:
    // lanes 0-15 hold K=0/1 in vgpr0/1, lanes 16-31 hold K=2/3).
    v2f aL[4], aT[4];
    {
        const int M  = lane & 15;
        const int kh = (lane < 16) ? 0 : 2;
#pragma unroll
        for (int g = 0; g < 4; ++g) {
            int K0 = 4*g + kh, K1 = K0 + 1;
            aL[g].x = (K0 <= M) ? 1.0f : 0.0f;
            aL[g].y = (K1 <= M) ? 1.0f : 0.0f;
            aT[g].x = (K0 <= M) ? (float)(M - K0 + 1) : 0.0f;
            aT[g].y = (K1 <= M) ? (float)(M - K1 + 1) : 0.0f;
        }
    }

    __syncthreads();

    for (int chunk = 0; chunk < NCHUNK; ++chunk) {
        const int par   = chunk & 1;
        const int cbase = chunk * CHUNK;
        const int j0    = cbase + tid * LSTEPS;

        if (chunk + 1 < NCHUNK) {   // pull next chunk toward L2 early
            __builtin_prefetch(predB + (size_t)(j0 + CHUNK) * 6, 0, 1);
            __builtin_prefetch(usB   + (size_t)(j0 + CHUNK) * 3, 0, 1);
            __builtin_prefetch(gtB   + (size_t)(j0 + CHUNK) * 10, 0, 1);
        }

        // ---- A1: per-thread serial chain of 8 delta-quaternions ----
        Quat lq[LSTEPS];
        Quat run; run.w = 1.f; run.x = 0.f; run.y = 0.f; run.z = 0.f;
#pragma unroll
        for (int k = 0; k < LSTEPS; ++k) {
            int j = j0 + k;
            Quat dq; dq.w = 1.f; dq.x = 0.f; dq.y = 0.f; dq.z = 0.f;
            if (j < NSTEP) {
                float gx = predB[(size_t)j*6 + 0];
                float gy = predB[(size_t)j*6 + 1];
                float gz = predB[(size_t)j*6 + 2];
                float ang  = sqrtf(gx*gx + gy*gy + gz*gz);
                float half = 0.5f * ang * DTg;
                float s = sinf(half), c = cosf(half);
                float inv = s / (ang + EPSg);
                dq.w = c; dq.x = gx*inv; dq.y = gy*inv; dq.z = gz*inv;
            }
            run = qnorm(qmul(run, dq));
            lq[k] = run;
        }

        // ---- A2: wave32 inclusive scan of thread products (non-commutative) ----
        Quat wp = run;
#pragma unroll
        for (int off = 1; off < 32; off <<= 1) {
            Quat p = qshfl_up(wp, off);
            if (lane >= off) wp = qnorm(qmul(p, wp));
        }
        Quat wpe = qshfl_up(wp, 1);
        if (lane == 0) { wpe.w = 1.f; wpe.x = 0.f; wpe.y = 0.f; wpe.z = 0.f; }
        if (lane == 31) {
            s_waveQ[wid][0] = wp.w; s_waveQ[wid][1] = wp.x;
            s_waveQ[wid][2] = wp.y; s_waveQ[wid][3] = wp.z;
        }
        __syncthreads();

        // ---- A3: thread 0 scans the 8 wave aggregates ----
        if (tid == 0) {
            Quat a; a.w = 1.f; a.x = 0.f; a.y = 0.f; a.z = 0.f;
            for (int w = 0; w < WAVES; ++w) {
                s_preQ[w][0] = a.w; s_preQ[w][1] = a.x;
                s_preQ[w][2] = a.y; s_preQ[w][3] = a.z;
                Quat ag; ag.w = s_waveQ[w][0]; ag.x = s_waveQ[w][1];
                ag.y = s_waveQ[w][2]; ag.z = s_waveQ[w][3];
                a = qnorm(qmul(a, ag));
            }
            Quat qc0; qc0.w = s_qc[par][0]; qc0.x = s_qc[par][1];
            qc0.y = s_qc[par][2]; qc0.z = s_qc[par][3];
            Quat nc = qnorm(qmul(qc0, a));
            s_qc[1-par][0] = nc.w; s_qc[1-par][1] = nc.x;
            s_qc[1-par][2] = nc.y; s_qc[1-par][3] = nc.z;
        }
        __syncthreads();

        // ---- A4: full attitude, NED accel, q-loss, wave v/p aggregates ----
        Quat qcar; qcar.w = s_qc[par][0]; qcar.x = s_qc[par][1];
        qcar.y = s_qc[par][2]; qcar.z = s_qc[par][3];
        Quat pre;  pre.w = s_preQ[wid][0]; pre.x = s_preQ[wid][1];
        pre.y = s_preQ[wid][2]; pre.z = s_preQ[wid][3];
        Quat E = qnorm(qmul(qmul(qcar, pre), wpe));

        float SvL[3] = {0.f, 0.f, 0.f}, SpL[3] = {0.f, 0.f, 0.f};
#pragma unroll
        for (int k = 0; k < LSTEPS; ++k) {
            int j = j0 + k;
            Quat q = qnorm(qmul(E, lq[k]));
            float ax = 0.f, ay = 0.f, az = 0.f;
            if (j < NSTEP) {
                int t = j + 1;
                const float* g4 = gtB + (size_t)t * 10;
                float dqw = q.w - g4[6], dqx = q.x - g4[7];
                float dqy = q.y - g4[8], dqz = q.z - g4[9];
                acc += (dqw*dqw + dqx*dqx + dqy*dqy + dqz*dqz) * invQ;
                float vx = usB[(size_t)j*3 + 0];
                float vy = usB[(size_t)j*3 + 1];
                float vz = usB[(size_t)j*3 + 2];
                // r = v + qw*t2 + cross(qv,t2), t2 = 2*cross(qv,v)
                float tx = 2.0f * (q.y*vz - q.z*vy);
                float ty = 2.0f * (q.z*vx - q.x*vz);
                float tz = 2.0f * (q.x*vy - q.y*vx);
                ax = vx + q.w*tx + (q.y*tz - q.z*ty);
                ay = vy + q.w*ty + (q.z*tx - q.x*tz);
                az = vz + q.w*tz + (q.x*ty - q.y*tx) - 9.81f;
            }
            int li = tid * LSTEPS + k;
            s_a[0][li] = ax; s_a[1][li] = ay; s_a[2][li] = az;
            float wgt = (float)(256 - (lane * LSTEPS + k));   // (n - i) multiplier
            SvL[0] += ax;      SvL[1] += ay;      SvL[2] += az;
            SpL[0] += wgt*ax;  SpL[1] += wgt*ay;  SpL[2] += wgt*az;
        }
#pragma unroll
        for (int off = 16; off >= 1; off >>= 1) {
#pragma unroll
            for (int c = 0; c < 3; ++c) {
                SvL[c] += __shfl_xor(SvL[c], off, 32);
                SpL[c] += __shfl_xor(SpL[c], off, 32);
            }
        }
        if (lane == 0)
            for (int c = 0; c < 3; ++c) { s_wSv[wid][c] = SvL[c]; s_wSp[wid][c] = SpL[c]; }
        __syncthreads();

        // ---- A5: thread 0 scans v/p wave aggregates (affine combine) ----
        if (tid == 0) {
            float vin[3], pin[3];
            for (int c = 0; c < 3; ++c) { vin[c] = s_vc[par][c]; pin[c] = s_pc[par][c]; }
            for (int w = 0; w < WAVES; ++w) {
                for (int c = 0; c < 3; ++c) { s_wVin[w][c] = vin[c]; s_wPin[w][c] = pin[c]; }
                for (int c = 0; c < 3; ++c) {
                    float pn = pin[c] + 256.0f*DTg*vin[c] + DT2g*s_wSp[w][c];
                    float vn = vin[c] + DTg*s_wSv[w][c];
                    pin[c] = pn; vin[c] = vn;
                }
            }
            for (int c = 0; c < 3; ++c) { s_vc[1-par][c] = vin[c]; s_pc[1-par][c] = pin[c]; }
        }
        __syncthreads();

        // ---- B: per-wave WMMA double-prefix over 16-step tiles ----
        {
            const int wbase = wid * 256;
            float vcol = (n < 3) ? s_wVin[wid][n] : 0.0f;
            float pcol = (n < 3) ? s_wPin[wid][n] : 0.0f;
            // every lane has a valid column pointer: n>=3 -> zero row
            const float* lanePtr = s_a[(n < 3) ? n : 3];
            const int kh = (lane < 16) ? 0 : 2;

            for (int tile = 0; tile < 16; ++tile) {
                const int tbase = wbase + tile * 16;

                // fetch all 4 K-slices up front: 4 unconditional ds_load_b64,
                // one wait, then 8 back-to-back WMMAs
                v2f bxs[4];
#pragma unroll
                for (int g = 0; g < 4; ++g) {
                    int K0 = 4*g + kh;                      // even -> 8B aligned
                    const float2 p2 = *(const float2*)(lanePtr + tbase + K0);
                    bxs[g].x = p2.x; bxs[g].y = p2.y;
                }

                v8f cl = {};   // sv = L * X   (inclusive prefix sums)
                v8f ct = {};   // sp = T * X   (weighted prefix sums)
#pragma unroll
                for (int g = 0; g < 4; ++g) {
                    cl = __builtin_amdgcn_wmma_f32_16x16x4_f32(
                        false, aL[g], false, bxs[g], (short)0, cl, false, false);
                    ct = __builtin_amdgcn_wmma_f32_16x16x4_f32(
                        false, aT[g], false, bxs[g], (short)0, ct, false, false);
                }

                // row 15 of D lives in vgpr 7 of lanes 16..31 (M = d+8)
                float Sv = __shfl(cl[7], 16 + n, 32);
                float Sp = __shfl(ct[7], 16 + n, 32);
                if (n < 3) {
#pragma unroll
                    for (int d = 0; d < 8; ++d) {
                        int m = d + ((lane < 16) ? 0 : 8);
                        int j = cbase + tbase + m;
                        if (j < NSTEP) {
                            float vt = vcol + DTg * cl[d];
                            float pt = pcol + (float)(m + 1) * DTg * vcol + DT2g * ct[d];
                            const float* gr = gtB + (size_t)(j + 1) * 10;
                            float dv = vt - gr[n];
                            float dp = pt - gr[3 + n];
                            acc += (dv*dv + dp*dp) * invVP;
                        }
                    }
                }
                pcol = pcol + 16.0f * DTg * vcol + DT2g * Sp;
                vcol = vcol + DTg * Sv;
            }
        }
        __syncthreads();
    }

    // ---- final reduction: wave32 shuffles -> LDS -> one float atomic ----
#pragma unroll
    for (int off = 16; off >= 1; off >>= 1) acc += __shfl_xor(acc, off, 32);
    if (lane == 0) s_red[wid] = acc;
    __syncthreads();
    if (wid == 0) {
        float v = (lane < WAVES) ? s_red[lane] : 0.0f;
#pragma unroll
        for (int off = 4; off >= 1; off >>= 1) v += __shfl_xor(v, off, 32);
        if (lane == 0) unsafeAtomicAdd(out, v);
    }
}

extern "C" void kernel_launch(void* const* d_in, const int* in_sizes, int n_in,
                              void* d_out, int out_size, void* d_ws, size_t ws_size,
                              hipStream_t stream) {
    const float* predicted    = (const float*)d_in[0];
    const float* ground_truth = (const float*)d_in[1];
    const float* us           = (const float*)d_in[2];
    float* out = (float*)d_out;
    (void)in_sizes; (void)n_in; (void)out_size; (void)d_ws; (void)ws_size;

    ins_zero_out_kernel<<<1, 32, 0, stream>>>(out);
    ins_loss_kernel<<<B_, BLOCK, 0, stream>>>(predicted, ground_truth, us, out);
}